// PathwayToCellGate_14027363189044
// MI455X (gfx1250) — compile-verified
//
#include <hip/hip_runtime.h>
#include <math.h>

typedef float v2f __attribute__((ext_vector_type(2)));
typedef float v8f __attribute__((ext_vector_type(8)));

#define DPC 128   // pathway/cell feature dim
#define DOUT 128  // output dim
#define EPS_SM 1e-16f
#define EPS_STD 1e-6f

__device__ __forceinline__ float sigmoidf_(float x) { return 1.0f / (1.0f + expf(-x)); }

// ordered-uint encoding of float for atomic max (monotonic map)
__device__ __forceinline__ unsigned fkey(float f) {
  unsigned b = __float_as_uint(f);
  return (b & 0x80000000u) ? ~b : (b | 0x80000000u);
}
__device__ __forceinline__ float funkey(unsigned u) {
  return __uint_as_float((u & 0x80000000u) ? (u & 0x7FFFFFFFu) : ~u);
}

// ------------------------------------------------------------------
// Kernel 1: pathway precompute (3 fused fp32 WMMA GEMMs, M=P,K=128,N=128)
//   A_p = pathway_h @ W_a1[0:128]
//   G_p = sigmoid(pathway_h @ W_g + b_g) * (pathway_h @ W_proj + b_proj)
// one wave -> one 16x16 output tile
// ------------------------------------------------------------------
__global__ void pathway_gemm(const float* __restrict__ ph,
                             const float* __restrict__ Wa1,
                             const float* __restrict__ Wg, const float* __restrict__ bg,
                             const float* __restrict__ Wp, const float* __restrict__ bp,
                             float* __restrict__ A_p, float* __restrict__ G_p, int P)
{
  int wid = (blockIdx.x * blockDim.x + threadIdx.x) >> 5;
  int nMt = P >> 4;
  if (wid >= nMt * (DOUT / 16)) return;
  int m0 = (wid >> 3) << 4;
  int n0 = (wid & 7) << 4;
  int lane = threadIdx.x & 31;
  int half = lane >> 4;     // 0: K0/K1 , 1: K2/K3
  int lm   = lane & 15;
  int col  = n0 + lm;

  v8f accA = {}; v8f accG = {}; v8f accP = {};
  for (int k = 0; k < DPC; k += 4) {
    int ka = k + half * 2;
    const float* ar = ph + (size_t)(m0 + lm) * DPC + ka;
    v2f a;  a.x  = ar[0];              a.y  = ar[1];
    v2f b1; b1.x = Wa1[ka * DOUT + col]; b1.y = Wa1[(ka + 1) * DOUT + col];
    v2f b2; b2.x = Wg [ka * DOUT + col]; b2.y = Wg [(ka + 1) * DOUT + col];
    v2f b3; b3.x = Wp [ka * DOUT + col]; b3.y = Wp [(ka + 1) * DOUT + col];
    accA = __builtin_amdgcn_wmma_f32_16x16x4_f32(false, a, false, b1, (short)0, accA, false, false);
    accG = __builtin_amdgcn_wmma_f32_16x16x4_f32(false, a, false, b2, (short)0, accG, false, false);
    accP = __builtin_amdgcn_wmma_f32_16x16x4_f32(false, a, false, b3, (short)0, accP, false, false);
  }
  float bgc = bg[col], bpc = bp[col];
  for (int r = 0; r < 8; ++r) {
    int row = m0 + r + half * 8;
    A_p[(size_t)row * DOUT + col] = accA[r];
    G_p[(size_t)row * DOUT + col] = sigmoidf_(accG[r] + bgc) * (accP[r] + bpc);
  }
}

// ------------------------------------------------------------------
// Kernel 2: A_c = cell_feat @ W_a1[128:256] + b_a1   (M=C,K=128,N=128)
// ------------------------------------------------------------------
__global__ void cell_gemm(const float* __restrict__ cf,
                          const float* __restrict__ Wa1bot,  // already offset to row 128
                          const float* __restrict__ ba1,
                          float* __restrict__ A_c, int C)
{
  int wid = (blockIdx.x * blockDim.x + threadIdx.x) >> 5;
  int nMt = C >> 4;
  if (wid >= nMt * (DOUT / 16)) return;
  int m0 = (wid >> 3) << 4;
  int n0 = (wid & 7) << 4;
  int lane = threadIdx.x & 31;
  int half = lane >> 4;
  int lm   = lane & 15;
  int col  = n0 + lm;

  v8f acc = {};
  for (int k = 0; k < DPC; k += 4) {
    int ka = k + half * 2;
    const float* ar = cf + (size_t)(m0 + lm) * DPC + ka;
    v2f a; a.x = ar[0]; a.y = ar[1];
    v2f b; b.x = Wa1bot[ka * DOUT + col]; b.y = Wa1bot[(ka + 1) * DOUT + col];
    acc = __builtin_amdgcn_wmma_f32_16x16x4_f32(false, a, false, b, (short)0, acc, false, false);
  }
  float bc = ba1[col];
  for (int r = 0; r < 8; ++r) {
    int row = m0 + r + half * 8;
    A_c[(size_t)row * DOUT + col] = acc[r] + bc;
  }
}

// ------------------------------------------------------------------
// Kernel 3: per-edge weight stats -> seg_sum / seg_sumsq / seg_cnt
// ------------------------------------------------------------------
__global__ void edge_stats(const float* __restrict__ ew, const int* __restrict__ dst,
                           float* __restrict__ ssum, float* __restrict__ ssq,
                           float* __restrict__ scnt, int E)
{
  int e = blockIdx.x * blockDim.x + threadIdx.x;
  if (e >= E) return;
  float w = ew[e];
  int d = dst[e];
  atomicAdd(&ssum[d], w);
  atomicAdd(&ssq[d], w * w);
  atomicAdd(&scnt[d], 1.0f);
}

// ------------------------------------------------------------------
// Kernel 4: per-cell mean / std  (var = E[x^2]-mean^2, clamp >= 0)
// ------------------------------------------------------------------
__global__ void cell_stats(float* __restrict__ ssum, float* __restrict__ ssq,
                           const float* __restrict__ scnt, int C)
{
  int c = blockIdx.x * blockDim.x + threadIdx.x;
  if (c >= C) return;
  float cnt  = fmaxf(scnt[c], 1.0f);
  float mean = ssum[c] / cnt;
  float var  = fmaxf(ssq[c] / cnt - mean * mean, 0.0f);
  ssum[c] = mean;                    // reuse: mean
  ssq[c]  = sqrtf(var + EPS_STD);    // reuse: std
}

// ------------------------------------------------------------------
// Kernel 5: per-edge attention logit (one wave per edge) + segment max
// att = tanh(A_p[src]+A_c[dst]) . W_a2 + b_a2 + alpha*(w-mean)/(std+eps)
// ------------------------------------------------------------------
__global__ void edge_att(const float* __restrict__ A_p, const float* __restrict__ A_c,
                         const float* __restrict__ Wa2, const float* __restrict__ ba2,
                         const float* __restrict__ ew, const float* __restrict__ alpha,
                         const int* __restrict__ src, const int* __restrict__ dst,
                         const float* __restrict__ mean, const float* __restrict__ std_,
                         float* __restrict__ att, unsigned* __restrict__ smax, int E)
{
  int e = (int)((blockIdx.x * (size_t)blockDim.x + threadIdx.x) >> 5);
  if (e >= E) return;
  int lane = threadIdx.x & 31;
  int s = src[e], d = dst[e];
  const float* pa = A_p + (size_t)s * DOUT;
  const float* ca = A_c + (size_t)d * DOUT;
  float acc = 0.0f;
  #pragma unroll
  for (int j = 0; j < DOUT / 32; ++j) {
    int idx = lane + j * 32;
    acc += tanhf(pa[idx] + ca[idx]) * Wa2[idx];
  }
  #pragma unroll
  for (int off = 16; off > 0; off >>= 1) acc += __shfl_xor(acc, off, 32);
  if (lane == 0) {
    float norm = (ew[e] - mean[d]) / (std_[d] + EPS_STD);
    float a = acc + ba2[0] + alpha[0] * norm;
    att[e] = a;
    atomicMax(&smax[d], fkey(a));
  }
}

// ------------------------------------------------------------------
// Kernel 6: exp(att - segmax) in place + segment denominator
// ------------------------------------------------------------------
__global__ void edge_exp(float* __restrict__ att, const unsigned* __restrict__ smax,
                         const int* __restrict__ dst, float* __restrict__ den, int E)
{
  int e = blockIdx.x * blockDim.x + threadIdx.x;
  if (e >= E) return;
  int d = dst[e];
  float ex = expf(att[e] - funkey(smax[d]));
  att[e] = ex;
  atomicAdd(&den[d], ex);
}

// ------------------------------------------------------------------
// Kernel 7: weighted scatter (one wave per edge)
// cell_h[dst] += G_p[src] * (exp / (den + eps))
// ------------------------------------------------------------------
__global__ void edge_scatter(const float* __restrict__ G_p, const float* __restrict__ exp_,
                             const float* __restrict__ den,
                             const int* __restrict__ src, const int* __restrict__ dst,
                             float* __restrict__ out, int E)
{
  int e = (int)((blockIdx.x * (size_t)blockDim.x + threadIdx.x) >> 5);
  if (e >= E) return;
  int lane = threadIdx.x & 31;
  int s = src[e], d = dst[e];
  float w = exp_[e] / (den[d] + EPS_SM);
  const float* g = G_p + (size_t)s * DOUT;
  float* o = out + (size_t)d * DOUT;
  #pragma unroll
  for (int j = 0; j < DOUT / 32; ++j) {
    int idx = lane + j * 32;
    atomicAdd(&o[idx], g[idx] * w);
  }
}

// ------------------------------------------------------------------
extern "C" void kernel_launch(void* const* d_in, const int* in_sizes, int n_in,
                              void* d_out, int out_size, void* d_ws, size_t ws_size,
                              hipStream_t stream) {
  const float* ph   = (const float*)d_in[0];
  const float* cf   = (const float*)d_in[1];
  const float* ew   = (const float*)d_in[2];
  const int*   src  = (const int*)d_in[3];
  const int*   dst  = (const int*)d_in[4];
  const float* Wp   = (const float*)d_in[5];
  const float* bp   = (const float*)d_in[6];
  const float* Wa1  = (const float*)d_in[7];
  const float* ba1  = (const float*)d_in[8];
  const float* Wa2  = (const float*)d_in[9];
  const float* ba2  = (const float*)d_in[10];
  const float* Wg   = (const float*)d_in[11];
  const float* bg   = (const float*)d_in[12];
  const float* alpha= (const float*)d_in[13];

  int P = in_sizes[0] / DPC;
  int C = in_sizes[1] / DPC;
  int E = in_sizes[2];

  // workspace layout (floats)
  float* ws = (float*)d_ws;
  size_t oAp  = 0;
  size_t oGp  = oAp + (size_t)P * DOUT;
  size_t oAc  = oGp + (size_t)P * DOUT;
  size_t oAtt = oAc + (size_t)C * DOUT;
  size_t oSum = oAtt + (size_t)E;
  size_t oSq  = oSum + (size_t)C;
  size_t oCnt = oSq  + (size_t)C;
  size_t oDen = oCnt + (size_t)C;
  size_t oMax = oDen + (size_t)C;   // uint keys

  float* A_p  = ws + oAp;
  float* G_p  = ws + oGp;
  float* A_c  = ws + oAc;
  float* att  = ws + oAtt;
  float* ssum = ws + oSum;
  float* ssq  = ws + oSq;
  float* scnt = ws + oCnt;
  float* sden = ws + oDen;
  unsigned* smax = (unsigned*)(ws + oMax);

  // zero: seg arrays (sum,sq,cnt,den,maxkey contiguous) and output accumulator
  hipMemsetAsync(ssum, 0, (size_t)5 * C * sizeof(float), stream);
  hipMemsetAsync(d_out, 0, (size_t)out_size * sizeof(float), stream);

  // 1) pathway precompute (WMMA f32)
  {
    int waves = (P / 16) * (DOUT / 16);
    int blocks = (waves + 3) / 4;                    // 4 waves / 128 threads per block
    pathway_gemm<<<blocks, 128, 0, stream>>>(ph, Wa1, Wg, bg, Wp, bp, A_p, G_p, P);
  }
  // 2) cell precompute (WMMA f32); W_a1 bottom half starts at row DPC
  {
    int waves = (C / 16) * (DOUT / 16);
    int blocks = (waves + 3) / 4;
    cell_gemm<<<blocks, 128, 0, stream>>>(cf, Wa1 + (size_t)DPC * DOUT, ba1, A_c, C);
  }
  // 3) edge weight stats
  edge_stats<<<(E + 255) / 256, 256, 0, stream>>>(ew, dst, ssum, ssq, scnt, E);
  // 4) per-cell mean/std
  cell_stats<<<(C + 255) / 256, 256, 0, stream>>>(ssum, ssq, scnt, C);
  // 5) per-edge attention logit + segment max (wave per edge)
  edge_att<<<(E + 7) / 8, 256, 0, stream>>>(A_p, A_c, Wa2, ba2, ew, alpha, src, dst,
                                            ssum, ssq, att, smax, E);
  // 6) exp + denominator
  edge_exp<<<(E + 255) / 256, 256, 0, stream>>>(att, smax, dst, sden, E);
  // 7) weighted scatter (wave per edge)
  edge_scatter<<<(E + 7) / 8, 256, 0, stream>>>(G_p, att, sden, src, dst, (float*)d_out, E);
}